// MultiLayerGCN_83141976916995
// MI455X (gfx1250) — compile-verified
//
#include <hip/hip_runtime.h>
#include <hip/hip_bf16.h>
#include <math.h>

#define BN_EPS 1e-5f

typedef __attribute__((ext_vector_type(2))) float v2f;
typedef __attribute__((ext_vector_type(8))) float v8f;

__device__ __forceinline__ float gelu_erf(float x) {
    return 0.5f * x * (1.0f + erff(x * 0.70710678118654752f));
}

// ---------------- utility ----------------
__global__ void fill_kernel(float* __restrict__ p, float v, int n) {
    int i = blockIdx.x * blockDim.x + threadIdx.x;
    if (i < n) p[i] = v;
}

// ---------------- degree / norm ----------------
__global__ void degree_kernel(const int* __restrict__ dst, float* __restrict__ deg, int E) {
    int e = blockIdx.x * blockDim.x + threadIdx.x;
    if (e < E) unsafeAtomicAdd(&deg[dst[e]], 1.0f);
}

__global__ void dinv_kernel(const float* __restrict__ deg, float* __restrict__ dinv, int n) {
    int i = blockIdx.x * blockDim.x + threadIdx.x;
    if (i < n) dinv[i] = rsqrtf(deg[i]);
}

// ---------------- fp32 WMMA GEMM: C[N,M] = A[N,K] @ B[K,M] (+bias) ----------------
// Block = 8 waves; B col-tile (K x 16) staged once in LDS (pair-interleaved so each
// lane's {B[k][c],B[k+1][c]} fragment is one conflict-free ds_load_b64). Each wave
// computes TWO 16x16 row tiles sharing each B fragment: 2 WMMAs per LDS read.
__global__ __launch_bounds__(256) void gemm_wmma_f32(
    const float* __restrict__ A, const float* __restrict__ B,
    float* __restrict__ C, const float* __restrict__ bias,
    int Nrows, int K, int M)
{
    __shared__ float sB[4096];                      // up to K=256: 256*16 floats = 16 KB

    const int lane = threadIdx.x & 31;
    const int wave = threadIdx.x >> 5;
    const int c0 = blockIdx.x << 4;

    // ---- stage B tile: element (k, j) -> sB[(k>>1)*32 + 2*j + (k&1)] ----
    for (int idx = threadIdx.x; idx < (K << 4); idx += 256) {
        int k = idx >> 4;
        int j = idx & 15;
        sB[((k >> 1) << 5) + (j << 1) + (k & 1)] = B[(size_t)k * M + c0 + j];
    }
    __syncthreads();

    const int nRowTiles = Nrows >> 4;
    const int rowTile0 = (blockIdx.y << 4) + (wave << 1);
    if (rowTile0 >= nRowTiles) return;              // uniform per wave: EXEC stays all-1s
    const int rowTile1 = rowTile0 + 1;
    const bool do1 = (rowTile1 < nRowTiles);

    const int half = lane >> 4;                     // 0: K pair {0,1}; 1: K pair {2,3}
    const int l = lane & 15;
    const int r0 = rowTile0 << 4;
    const int r1 = do1 ? (rowTile1 << 4) : r0;      // clamp loads when tile1 invalid

    v8f acc0 = {0.f, 0.f, 0.f, 0.f, 0.f, 0.f, 0.f, 0.f};
    v8f acc1 = {0.f, 0.f, 0.f, 0.f, 0.f, 0.f, 0.f, 0.f};

    // A (16x4, 32-bit layout): lanes 0-15 row=l K={k,k+1}; lanes 16-31 row=l K={k+2,k+3}
    const float* arow0 = A + (size_t)(r0 + l) * K + 2 * half;
    const float* arow1 = A + (size_t)(r1 + l) * K + 2 * half;
    const float* bfrag = sB + (half << 5) + (l << 1);

    for (int k = 0; k < K; k += 4) {
        v2f a0 = *(const v2f*)(arow0 + k);
        v2f a1 = *(const v2f*)(arow1 + k);
        v2f b  = *(const v2f*)(bfrag + (k << 3)); // ((k>>2)+half)*32 + 2l, k multiple of 4
        acc0 = __builtin_amdgcn_wmma_f32_16x16x4_f32(
            false, a0, false, b, (short)0, acc0, false, false);
        acc1 = __builtin_amdgcn_wmma_f32_16x16x4_f32(
            false, a1, false, b, (short)0, acc1, false, false);
    }

    // C/D layout: VGPR j -> row +j (lanes 0-15) / +8+j (lanes 16-31), col = c0+l
    const int col = c0 + l;
    const float badd = bias ? bias[col] : 0.0f;
    float* crow0 = C + (size_t)(r0 + half * 8) * M + col;
#pragma unroll
    for (int j = 0; j < 8; ++j) crow0[(size_t)j * M] = acc0[j] + badd;
    if (do1) {
        float* crow1 = C + (size_t)(r1 + half * 8) * M + col;
#pragma unroll
        for (int j = 0; j < 8; ++j) crow1[(size_t)j * M] = acc1[j] + badd;
    }
}

// ---------------- GCN aggregation ----------------
// agg = self_norm * xw + bias
__global__ void agg_init_kernel(const float* __restrict__ xw, const float* __restrict__ dinv,
                                const float* __restrict__ bias, float* __restrict__ agg,
                                int Nn, int F) {
    int idx = blockIdx.x * blockDim.x + threadIdx.x;
    int total = Nn * F;
    if (idx >= total) return;
    int n = idx / F;
    int f = idx - n * F;
    float dn = dinv[n];
    agg[idx] = dn * dn * xw[idx] + bias[f];
}

// agg[dst] += dinv[src]*dinv[dst] * xw[src]   (one thread per edge x 4-feature group)
__global__ void edge_scatter_kernel(const int* __restrict__ ei, const float* __restrict__ dinv,
                                    const float* __restrict__ xw, float* __restrict__ agg,
                                    int E, int F) {
    const int ng = F >> 2;
    long long idx = (long long)blockIdx.x * blockDim.x + threadIdx.x;
    long long total = (long long)E * ng;
    if (idx >= total) return;
    int e = (int)(idx / ng);
    int g = ((int)(idx - (long long)e * ng)) << 2;
    int s = ei[e];
    int d = ei[E + e];
    float nrm = dinv[s] * dinv[d];
    const float4 v = *(const float4*)(xw + (size_t)s * F + g);
    float* ap = agg + (size_t)d * F + g;
    unsafeAtomicAdd(ap + 0, nrm * v.x);
    unsafeAtomicAdd(ap + 1, nrm * v.y);
    unsafeAtomicAdd(ap + 2, nrm * v.z);
    unsafeAtomicAdd(ap + 3, nrm * v.w);
}

// ---------------- batch norm (training-mode batch stats over nodes) ----------------
__global__ void bn_stats_kernel(const float* __restrict__ X, float* __restrict__ sums,
                                int Nn, int F, int rowsPerBlock) {
    int f = threadIdx.x;
    if (f >= F) return;
    int r0 = blockIdx.x * rowsPerBlock;
    int r1 = r0 + rowsPerBlock;
    if (r1 > Nn) r1 = Nn;
    float s = 0.f, s2 = 0.f;
    for (int r = r0; r < r1; ++r) {
        float v = X[(size_t)r * F + f];
        s += v;
        s2 += v * v;
    }
    unsafeAtomicAdd(&sums[f], s);
    unsafeAtomicAdd(&sums[F + f], s2);
}

__global__ void bn_finalize_kernel(const float* __restrict__ sums, const float* __restrict__ g,
                                   const float* __restrict__ b, float* __restrict__ ss,
                                   int Nn, int F) {
    int f = blockIdx.x * blockDim.x + threadIdx.x;
    if (f >= F) return;
    float inv_n = 1.0f / (float)Nn;
    float mean = sums[f] * inv_n;
    float var = sums[F + f] * inv_n - mean * mean;   // biased variance (matches jnp.var)
    float scale = g[f] * rsqrtf(var + BN_EPS);
    ss[f] = scale;
    ss[F + f] = b[f] - mean * scale;
}

__global__ void bn_apply_gelu_kernel(float* __restrict__ X, const float* __restrict__ ss,
                                     int Nn, int F) {
    int idx = blockIdx.x * blockDim.x + threadIdx.x;
    int total = Nn * F;
    if (idx >= total) return;
    int f = idx % F;
    float v = X[idx] * ss[f] + ss[F + f];
    X[idx] = gelu_erf(v);
}

// ---------------- layer norm: one wave32 per row ----------------
__global__ __launch_bounds__(256) void layernorm_kernel(const float* __restrict__ X,
                                                        float* __restrict__ Y,
                                                        const float* __restrict__ g,
                                                        const float* __restrict__ b,
                                                        int Nn, int F) {
    const int lane = threadIdx.x & 31;
    const int wave = threadIdx.x >> 5;
    const int row = blockIdx.x * 8 + wave;
    if (row >= Nn) return;
    const float* x = X + (size_t)row * F;
    float s = 0.f, s2 = 0.f;
    for (int f = lane; f < F; f += 32) {
        float v = x[f];
        s += v;
        s2 += v * v;
    }
#pragma unroll
    for (int off = 16; off > 0; off >>= 1) {
        s  += __shfl_xor(s,  off, 32);
        s2 += __shfl_xor(s2, off, 32);
    }
    float inv_f = 1.0f / (float)F;
    float mean = s * inv_f;
    float var = s2 * inv_f - mean * mean;
    float rstd = rsqrtf(var + BN_EPS);
    float* y = Y + (size_t)row * F;
    for (int f = lane; f < F; f += 32) {
        y[f] = (x[f] - mean) * rstd * g[f] + b[f];
    }
}

// ---------------- fused predictor MLP: 16 -> 32 -> 16 -> 8 -> 1 ----------------
__global__ __launch_bounds__(256) void predictor_kernel(
    const float* __restrict__ H,
    const float* __restrict__ W1, const float* __restrict__ b1,
    const float* __restrict__ g1, const float* __restrict__ be1,
    const float* __restrict__ W2, const float* __restrict__ b2,
    const float* __restrict__ g2, const float* __restrict__ be2,
    const float* __restrict__ W3, const float* __restrict__ b3,
    const float* __restrict__ W4, const float* __restrict__ b4,
    float* __restrict__ out, int Nn)
{
    __shared__ float sW1[512], sW2[512], sW3[128];
    __shared__ float sb1[32], sg1[32], sbe1[32];
    __shared__ float sb2[16], sg2[16], sbe2[16];
    __shared__ float sb3[8], sW4[8];
    __shared__ float sb4;
    const int t = threadIdx.x;
    for (int i = t; i < 512; i += 256) { sW1[i] = W1[i]; sW2[i] = W2[i]; }
    if (t < 128) sW3[t] = W3[t];
    if (t < 32) { sb1[t] = b1[t]; sg1[t] = g1[t]; sbe1[t] = be1[t]; }
    if (t < 16) { sb2[t] = b2[t]; sg2[t] = g2[t]; sbe2[t] = be2[t]; }
    if (t < 8)  { sb3[t] = b3[t]; sW4[t] = W4[t]; }
    if (t == 0) sb4 = b4[0];
    __syncthreads();

    int n = blockIdx.x * blockDim.x + t;
    if (n >= Nn) return;

    float h0[16];
    const float* hp = H + (size_t)n * 16;
#pragma unroll
    for (int k = 0; k < 16; ++k) h0[k] = hp[k];

    // layer 1: 16 -> 32, LN, gelu
    float h1[32];
#pragma unroll
    for (int j = 0; j < 32; ++j) {
        float s = sb1[j];
#pragma unroll
        for (int k = 0; k < 16; ++k) s += h0[k] * sW1[k * 32 + j];
        h1[j] = s;
    }
    {
        float mu = 0.f;
#pragma unroll
        for (int j = 0; j < 32; ++j) mu += h1[j];
        mu *= (1.0f / 32.0f);
        float var = 0.f;
#pragma unroll
        for (int j = 0; j < 32; ++j) { float d = h1[j] - mu; var += d * d; }
        var *= (1.0f / 32.0f);
        float rs = rsqrtf(var + BN_EPS);
#pragma unroll
        for (int j = 0; j < 32; ++j) h1[j] = gelu_erf((h1[j] - mu) * rs * sg1[j] + sbe1[j]);
    }

    // layer 2: 32 -> 16, LN, gelu
    float h2[16];
#pragma unroll
    for (int j = 0; j < 16; ++j) {
        float s = sb2[j];
#pragma unroll
        for (int k = 0; k < 32; ++k) s += h1[k] * sW2[k * 16 + j];
        h2[j] = s;
    }
    {
        float mu = 0.f;
#pragma unroll
        for (int j = 0; j < 16; ++j) mu += h2[j];
        mu *= (1.0f / 16.0f);
        float var = 0.f;
#pragma unroll
        for (int j = 0; j < 16; ++j) { float d = h2[j] - mu; var += d * d; }
        var *= (1.0f / 16.0f);
        float rs = rsqrtf(var + BN_EPS);
#pragma unroll
        for (int j = 0; j < 16; ++j) h2[j] = gelu_erf((h2[j] - mu) * rs * sg2[j] + sbe2[j]);
    }

    // layer 3: 16 -> 8, gelu
    float h3[8];
#pragma unroll
    for (int j = 0; j < 8; ++j) {
        float s = sb3[j];
#pragma unroll
        for (int k = 0; k < 16; ++k) s += h2[k] * sW3[k * 8 + j];
        h3[j] = gelu_erf(s);
    }

    // layer 4: 8 -> 1
    float o = sb4;
#pragma unroll
    for (int k = 0; k < 8; ++k) o += h3[k] * sW4[k];
    out[n] = o;
}

// ---------------- host orchestration ----------------
extern "C" void kernel_launch(void* const* d_in, const int* in_sizes, int n_in,
                              void* d_out, int out_size, void* d_ws, size_t ws_size,
                              hipStream_t stream) {
    (void)in_sizes; (void)n_in; (void)out_size; (void)ws_size;
    const int Nn = 50000;
    const int E = 800000;
    const int T = 256;

    const float* x     = (const float*)d_in[0];
    const int*   ei    = (const int*)d_in[1];       // [2][E], src then dst
    const float* in_W  = (const float*)d_in[2];
    const float* in_b  = (const float*)d_in[3];
    const float* bn0_g = (const float*)d_in[4];
    const float* bn0_b = (const float*)d_in[5];
    const float* convW[5]; const float* convB[5];
    const float* bnG[5]; const float* bnB[5];
    const float* lnG[5]; const float* lnB[5];
    for (int i = 0; i < 5; ++i) {
        convW[i] = (const float*)d_in[6 + i];
        convB[i] = (const float*)d_in[11 + i];
        bnG[i]   = (const float*)d_in[16 + i];
        bnB[i]   = (const float*)d_in[21 + i];
        lnG[i]   = (const float*)d_in[26 + i];
        lnB[i]   = (const float*)d_in[31 + i];
    }
    const float* pW1  = (const float*)d_in[36];
    const float* pb1  = (const float*)d_in[37];
    const float* pg1  = (const float*)d_in[38];
    const float* pbe1 = (const float*)d_in[39];
    const float* pW2  = (const float*)d_in[40];
    const float* pb2  = (const float*)d_in[41];
    const float* pg2  = (const float*)d_in[42];
    const float* pbe2 = (const float*)d_in[43];
    const float* pW3  = (const float*)d_in[44];
    const float* pb3  = (const float*)d_in[45];
    const float* pW4  = (const float*)d_in[46];
    const float* pb4  = (const float*)d_in[47];

    float* ws = (float*)d_ws;
    size_t off = 0;
    float* bufA = ws + off; off += (size_t)Nn * 256;   // h
    float* bufB = ws + off; off += (size_t)Nn * 256;   // xw
    float* bufC = ws + off; off += (size_t)Nn * 256;   // agg
    float* deg  = ws + off; off += Nn;
    float* dinv = ws + off; off += Nn;
    float* sums = ws + off; off += 512;                // [sum | sumsq]
    float* ss   = ws + off; off += 512;                // [scale | shift]

    // ---- degrees / symmetric normalization ----
    fill_kernel<<<(Nn + T - 1) / T, T, 0, stream>>>(deg, 1.0f, Nn);
    degree_kernel<<<(E + T - 1) / T, T, 0, stream>>>(ei + E, deg, E);
    dinv_kernel<<<(Nn + T - 1) / T, T, 0, stream>>>(deg, dinv, Nn);

    auto run_bn_gelu = [&](float* X, const float* g, const float* b, int F) {
        fill_kernel<<<(2 * F + T - 1) / T, T, 0, stream>>>(sums, 0.0f, 2 * F);
        const int rowsPerBlock = 128;
        const int nb = (Nn + rowsPerBlock - 1) / rowsPerBlock;
        const int bt = (F < 32) ? 32 : F;
        bn_stats_kernel<<<nb, bt, 0, stream>>>(X, sums, Nn, F, rowsPerBlock);
        bn_finalize_kernel<<<1, bt, 0, stream>>>(sums, g, b, ss, Nn, F);
        const int total = Nn * F;
        bn_apply_gelu_kernel<<<(total + T - 1) / T, T, 0, stream>>>(X, ss, Nn, F);
    };

    const int nRowTiles = Nn / 16;                     // 3125
    const int gy = (nRowTiles + 15) / 16;              // 196 (2 row tiles per wave, 8 waves)

    // ---- input projection 128 -> 256 (+bias), BN, gelu ----
    {
        dim3 grid(256 / 16, gy);
        gemm_wmma_f32<<<grid, 256, 0, stream>>>(x, in_W, bufA, in_b, Nn, 128, 256);
    }
    run_bn_gelu(bufA, bn0_g, bn0_b, 256);

    // ---- 5 GCN conv layers ----
    const int dims[6] = {256, 256, 128, 64, 32, 16};
    for (int i = 0; i < 5; ++i) {
        const int K = dims[i];
        const int F = dims[i + 1];
        // xw = h @ W
        dim3 grid(F / 16, gy);
        gemm_wmma_f32<<<grid, 256, 0, stream>>>(bufA, convW[i], bufB, nullptr, Nn, K, F);
        // agg = self_norm*xw + bias, then scatter edges
        const int total = Nn * F;
        agg_init_kernel<<<(total + T - 1) / T, T, 0, stream>>>(bufB, dinv, convB[i], bufC, Nn, F);
        const long long etotal = (long long)E * (F >> 2);
        edge_scatter_kernel<<<(int)((etotal + T - 1) / T), T, 0, stream>>>(ei, dinv, bufB, bufC, E, F);
        // BN + gelu (in place), then LN -> bufA (residual never fires: dims strictly decrease)
        run_bn_gelu(bufC, bnG[i], bnB[i], F);
        layernorm_kernel<<<(Nn + 7) / 8, 256, 0, stream>>>(bufC, bufA, lnG[i], lnB[i], Nn, F);
    }

    // ---- predictor head ----
    predictor_kernel<<<(Nn + T - 1) / T, T, 0, stream>>>(
        bufA, pW1, pb1, pg1, pbe1, pW2, pb2, pg2, pbe2, pW3, pb3, pW4, pb4,
        (float*)d_out, Nn);
}